// RayPointRefiner_3496103379245
// MI455X (gfx1250) — compile-verified
//
#include <hip/hip_runtime.h>
#include <stdint.h>

// RayPointRefiner for MI455X (gfx1250, wave32).
// Wave-per-ray, fully wave-synchronous (no block barriers needed: LDS ops are
// in-order within a wave). Async global->LDS staging of the depth row uses the
// CDNA5 ASYNCcnt path. Memory-bound: ~1KB traffic/ray -> ~22us floor @23.3TB/s.

#define RAYS_PER_BLOCK 8
#define PCOARSE 64   // coarse samples per ray
#define NSAMP   64   // importance samples per ray
#define NOUT    128  // output samples per ray
#define EPSV    1e-5f

__global__ __launch_bounds__(256) void ray_refine_kernel(
    const float* __restrict__ lengths,   // (R, 64) sorted depths
    const float* __restrict__ weights,   // (R, 64)
    float* __restrict__ out,             // (R, 128) sorted
    int R)
{
    const int lane = threadIdx.x & 31;
    const int wid  = threadIdx.x >> 5;
    const int ray  = blockIdx.x * RAYS_PER_BLOCK + wid;
    if (ray >= R) return;   // whole wave exits together; no block barriers used

    __shared__ __align__(16) float zbuf[RAYS_PER_BLOCK][PCOARSE];  // depths
    __shared__ __align__(16) float binbuf[RAYS_PER_BLOCK][PCOARSE]; // 63 mids
    __shared__ __align__(16) float cdfbuf[RAYS_PER_BLOCK][PCOARSE]; // 63 cdf
    __shared__ __align__(16) float sbuf[RAYS_PER_BLOCK][NSAMP];     // samples

    const float* zp = lengths + (size_t)ray * PCOARSE;
    const float* wp = weights + (size_t)ray * PCOARSE;

    // ---- Stage depth row into LDS via CDNA5 async copy (b64 per lane:
    //      32 lanes x 8B = one contiguous 256B row). Tracked on ASYNCcnt. ----
    {
        uint32_t ldsoff = (uint32_t)(uintptr_t)(&zbuf[wid][2 * lane]); // low 32b of flat ptr == LDS offset
        uint64_t gaddr  = (uint64_t)(uintptr_t)(zp + 2 * lane);
        asm volatile("global_load_async_to_lds_b64 %0, %1, off"
                     :: "v"(ldsoff), "v"(gaddr) : "memory");
    }

    // Plain coalesced load of the weight pair while the async copy is in flight.
    float2 wv = reinterpret_cast<const float2*>(wp)[lane];

    asm volatile("s_wait_asynccnt 0" ::: "memory");

    // Read depth pair back from LDS (ds_load_b64).
    float2 zv = *reinterpret_cast<const float2*>(&zbuf[wid][2 * lane]);

    // ---- bins = midpoints: mid[j] = 0.5*(z[j]+z[j+1]), j=0..62 ----
    float znext = __shfl_down(zv.x, 1, 32);        // z[2l+2] (junk on lane 31)
    binbuf[wid][2 * lane] = 0.5f * (zv.x + zv.y);  // mid[2l], 2l<=62 for all lanes
    if (lane <= 30)
        binbuf[wid][2 * lane + 1] = 0.5f * (zv.y + znext); // mid[2l+1]

    // ---- CDF over interior weights: slots 0..61 = w[1..62] + EPS ----
    float wnext = __shfl_down(wv.x, 1, 32);        // w[2l+2]
    float e0 = wv.y + EPSV;                        // slot 2l   <- w[2l+1]
    float e1 = wnext + EPSV;                       // slot 2l+1 <- w[2l+2]
    if (lane == 31) { e0 = 0.0f; e1 = 0.0f; }      // zero padding (slots 62,63)

    // Wave32 inclusive scan of per-lane pair sums (5 shfl_up steps).
    float pair = e0 + e1;
    float x = pair;
    #pragma unroll
    for (int d = 1; d < 32; d <<= 1) {
        float y = __shfl_up(x, d, 32);
        if (lane >= d) x += y;
    }
    float total = __shfl(x, 31, 32);               // S = sum of all 62 weights
    float excl  = x - pair;                        // exclusive prefix for slot 2l
    float invS  = 1.0f / total;

    if (lane == 0) cdfbuf[wid][0] = 0.0f;
    if (lane <= 30) {                              // cdf[k] = cumsum up to slot k-1
        cdfbuf[wid][2 * lane + 1] = (excl + e0) * invS;
        cdfbuf[wid][2 * lane + 2] = (excl + e0 + e1) * invS;
    }

    // ---- Inverse-CDF sampling: 2 samples per lane, searchsorted(side='right') ----
    #pragma unroll
    for (int q = 0; q < 2; ++q) {
        int s = lane + 32 * q;
        float u = (float)s * (1.0f / 63.0f);       // linspace(0,1,64)
        int lo = 0, hi = 63;                       // insertion point in [0,63]
        while (lo < hi) {                          // 6-step binary search in LDS
            int m = (lo + hi) >> 1;
            if (cdfbuf[wid][m] <= u) lo = m + 1; else hi = m;
        }
        int below = lo - 1; below = below < 0 ? 0 : (below > 62 ? 62 : below);
        int above = lo > 62 ? 62 : lo;
        float c0 = cdfbuf[wid][below], c1 = cdfbuf[wid][above];
        float b0 = binbuf[wid][below], b1 = binbuf[wid][above];
        float denom = c1 - c0;
        if (denom < EPSV) denom = 1.0f;
        float t = (u - c0) / denom;
        sbuf[wid][s] = b0 + t * (b1 - b0);         // monotone in s => sorted
    }

    // ---- Merge two sorted 64-lists (depths + samples) via merge-path.
    //      4 outputs per lane; full sort is unnecessary. ----
    const float* A = zbuf[wid];
    const float* B = sbuf[wid];
    float* op = out + (size_t)ray * NOUT;
    #pragma unroll
    for (int q = 0; q < 4; ++q) {
        int k = lane + 32 * q;
        int lo = k - NSAMP; if (lo < 0) lo = 0;
        int hi = k; if (hi > PCOARSE) hi = PCOARSE;
        while (lo < hi) {                          // diagonal binary search
            int m = (lo + hi) >> 1;
            if (A[m] <= B[k - 1 - m]) lo = m + 1; else hi = m;
        }
        int i = lo, j = k - lo;
        float a = (i < PCOARSE) ? A[i] : 3.0e38f;
        float b = (j < NSAMP)   ? B[j] : 3.0e38f;
        op[k] = fminf(a, b);
    }
}

extern "C" void kernel_launch(void* const* d_in, const int* in_sizes, int n_in,
                              void* d_out, int out_size, void* d_ws, size_t ws_size,
                              hipStream_t stream) {
    const float* lengths = (const float*)d_in[0];
    const float* rweights = (const float*)d_in[1];
    float* out = (float*)d_out;
    const int R = in_sizes[0] / PCOARSE;           // 524288
    const int blocks = (R + RAYS_PER_BLOCK - 1) / RAYS_PER_BLOCK;
    ray_refine_kernel<<<blocks, 256, 0, stream>>>(lengths, rweights, out, R);
}